// GCommNetMLP_34548716929128
// MI455X (gfx1250) — compile-verified
//
#include <hip/hip_runtime.h>
#include <hip/hip_bf16.h>

// ---------------------------------------------------------------------------
// GCommNet step for MI455X (gfx1250): fp32 WMMA (V_WMMA_F32_16X16X4_F32) for
// all GEMM stages; fused pairwise-adjacency MLP (avoids a 134MB intermediate).
// N=512 agents, H=256, NIN=128, 4 GAT heads of 64, P=2 passes.
// Round 2: branch-free templated GEMM, pointer-increment addressing,
// K unrolled x4 (4 WMMAs per loop body) for a denser wmma/valu mix.
// ---------------------------------------------------------------------------

typedef __attribute__((ext_vector_type(2))) float v2f;
typedef __attribute__((ext_vector_type(8))) float v8f;

#define NA    512   // agents
#define HD    256   // hidden
#define NINP  128
#define HEADS 4
#define HDH   64    // per-head dim
#define PASS  2

__device__ __forceinline__ float sigm_(float x) { return 1.0f / (1.0f + expf(-x)); }

// ---------------------------------------------------------------------------
// fp32 WMMA GEMM: C[M,N] = (ACC? C : 0) + A[M,K] @ op(B) + bias, ACT.
// One wave computes one 16x16 tile of C.  ACT: 0=none, 2=ELU.
// Fragment layouts per CDNA5 ISA 7.12.2 (16x4 f32 A; row-striped B/C).
// K must be a multiple of 16.
// ---------------------------------------------------------------------------
template <int TRANSB, int ACC, int ACT>
__global__ __launch_bounds__(256) void gemm_wmma_f32(
    const float* __restrict__ A, int lda,
    const float* __restrict__ B, int ldb,
    const float* __restrict__ bias,
    float* __restrict__ C, int ldc,
    int M, int N, int K)
{
    const int wave = (blockIdx.x * blockDim.x + threadIdx.x) >> 5;
    const int lane = threadIdx.x & 31;
    const int ntiles = N >> 4;
    const int mt = wave / ntiles;
    const int nt = wave - mt * ntiles;
    if (mt * 16 >= M) return;                 // wave-uniform: EXEC stays all-1s

    const int half = lane >> 4;
    const int lm   = lane & 15;
    const int arow = mt * 16 + lm;            // A-fragment row for this lane
    const int col  = nt * 16 + lm;            // B/C column for this lane

    // lane-private streaming pointers (K advances by +16 per iteration)
    const float* pA = A + (size_t)arow * lda + 2 * half;
    const float* pB = TRANSB ? (B + (size_t)col * ldb + 2 * half)
                             : (B + (size_t)(2 * half) * ldb + col);
    float* pC = C + (size_t)(mt * 16 + 8 * half) * ldc + col;

    v8f acc = {0.f, 0.f, 0.f, 0.f, 0.f, 0.f, 0.f, 0.f};
    if (ACC) {
#pragma unroll
        for (int v = 0; v < 8; ++v)
            acc[v] = pC[(size_t)v * ldc];
    }

    for (int k0 = 0; k0 < K; k0 += 16) {
        v2f a[4], b[4];
#pragma unroll
        for (int u = 0; u < 4; ++u) {         // 4 contiguous-pair loads (b64)
            a[u][0] = pA[u * 4 + 0];
            a[u][1] = pA[u * 4 + 1];
        }
#pragma unroll
        for (int u = 0; u < 4; ++u) {
            if (TRANSB) {                     // op(B)[k][n] = B[n*ldb + k]
                b[u][0] = pB[u * 4 + 0];
                b[u][1] = pB[u * 4 + 1];
            } else {                          // op(B)[k][n] = B[k*ldb + n]
                b[u][0] = pB[(size_t)(u * 4 + 0) * ldb];
                b[u][1] = pB[(size_t)(u * 4 + 1) * ldb];
            }
        }
#pragma unroll
        for (int u = 0; u < 4; ++u)
            acc = __builtin_amdgcn_wmma_f32_16x16x4_f32(
                false, a[u], false, b[u], (short)0, acc, false, false);
        pA += 16;
        pB += TRANSB ? (size_t)16 : (size_t)16 * ldb;
    }

    const float bv = bias ? bias[col] : 0.f;
#pragma unroll
    for (int v = 0; v < 8; ++v) {
        float x = acc[v] + bv;
        if (ACT == 2) x = (x > 0.f) ? x : (expf(x) - 1.f);   // ELU
        pC[(size_t)v * ldc] = x;
    }
}

static inline void launch_gemm(const float* A, int lda, const float* B, int ldb,
                               int transB, const float* bias, float* C, int ldc,
                               int M, int N, int K, int accum, int act,
                               hipStream_t s)
{
    int waves   = (M / 16) * (N / 16);
    int blocks  = (waves * 32 + 255) / 256;     // 8 waves / block (wave32)
#define GO(TB, AC, AT) gemm_wmma_f32<TB, AC, AT><<<blocks, 256, 0, s>>>( \
        A, lda, B, ldb, bias, C, ldc, M, N, K)
    if (transB) { if (accum) GO(1, 1, 0); else GO(1, 0, 0); }
    else if (act == 2)       GO(0, 0, 2);
    else if (accum)          GO(0, 1, 0);
    else                     GO(0, 0, 0);
#undef GO
}

// ---------------------------------------------------------------------------
// Fused pairwise adjacency:  z1 = relu(U[i]+V[j]+b1) (built on the fly),
// z2 = relu(z1@W2+b2) via two 16x16x4 WMMA accumulators (32 cols),
// logits = z2@W3+b3 via shfl_xor reduce, adj = hard-gumbel argmax bit.
// One wave = one (i, 16 j's) strip; 512*32 = 16384 waves.
// K=128 unrolled x4 -> 8 WMMAs per loop body.
// ---------------------------------------------------------------------------
__global__ __launch_bounds__(256) void pair_adj_kernel(
    const float* __restrict__ U,
    const float* __restrict__ V,
    const float* __restrict__ b1,
    const float* __restrict__ W2,
    const float* __restrict__ b2,
    const float* __restrict__ W3,
    const float* __restrict__ b3,
    const float* __restrict__ gu,
    float* __restrict__ adj)
{
    const int task = (blockIdx.x * blockDim.x + threadIdx.x) >> 5;
    const int lane = threadIdx.x & 31;
    const int JT   = NA / 16;
    const int irow = task / JT;
    const int j0   = (task - irow * JT) * 16;
    if (irow >= NA) return;

    const int half = lane >> 4;
    const int lm   = lane & 15;
    const float* pU = U + (size_t)irow * 128 + 2 * half;
    const float* pV = V + (size_t)(j0 + lm) * 128 + 2 * half;
    const float* pb = b1 + 2 * half;
    const float* pW = W2 + (size_t)(2 * half) * 32 + lm;   // +16 for 2nd tile

    v8f acc0 = {0.f,0.f,0.f,0.f,0.f,0.f,0.f,0.f};
    v8f acc1 = {0.f,0.f,0.f,0.f,0.f,0.f,0.f,0.f};

    for (int k0 = 0; k0 < 128; k0 += 16) {
        v2f a[4], bA[4], bB[4];
#pragma unroll
        for (int u = 0; u < 4; ++u) {
            a[u][0] = fmaxf(pU[u*4 + 0] + pV[u*4 + 0] + pb[u*4 + 0], 0.f);
            a[u][1] = fmaxf(pU[u*4 + 1] + pV[u*4 + 1] + pb[u*4 + 1], 0.f);
            bA[u][0] = pW[(size_t)(u*4 + 0) * 32];
            bA[u][1] = pW[(size_t)(u*4 + 1) * 32];
            bB[u][0] = pW[(size_t)(u*4 + 0) * 32 + 16];
            bB[u][1] = pW[(size_t)(u*4 + 1) * 32 + 16];
        }
#pragma unroll
        for (int u = 0; u < 4; ++u) {
            acc0 = __builtin_amdgcn_wmma_f32_16x16x4_f32(
                false, a[u], false, bA[u], (short)0, acc0, false, false);
            acc1 = __builtin_amdgcn_wmma_f32_16x16x4_f32(
                false, a[u], false, bB[u], (short)0, acc1, false, false);
        }
        pU += 16; pV += 16; pb += 16; pW += (size_t)16 * 32;
    }

    // z2 (16x32, C-layout) -> per-lane partial logits, reduce across 16 lanes.
    const float w30a = W3[lm * 2 + 0],        w31a = W3[lm * 2 + 1];
    const float w30b = W3[(16 + lm) * 2 + 0], w31b = W3[(16 + lm) * 2 + 1];
    const float b2a = b2[lm], b2b = b2[16 + lm];
    float p0[8], p1[8];
#pragma unroll
    for (int v = 0; v < 8; ++v) {
        float za = fmaxf(acc0[v] + b2a, 0.f);
        float zb = fmaxf(acc1[v] + b2b, 0.f);
        p0[v] = za * w30a + zb * w30b;
        p1[v] = za * w31a + zb * w31b;
    }
#pragma unroll
    for (int mask = 1; mask < 16; mask <<= 1) {   // stays within each 16-lane half
#pragma unroll
        for (int v = 0; v < 8; ++v) {
            p0[v] += __shfl_xor(p0[v], mask, 32);
            p1[v] += __shfl_xor(p1[v], mask, 32);
        }
    }
    if (lm == 0) {
#pragma unroll
        for (int v = 0; v < 8; ++v) {
            int j = j0 + v + 8 * half;            // C row -> pair index
            long idx = (long)irow * NA + j;
            float u0 = gu[idx * 2 + 0], u1 = gu[idx * 2 + 1];
            float g0 = -logf(-logf(fmaxf(fminf(u0, 1.f), 1e-10f)) + 1e-10f);
            float g1 = -logf(-logf(fmaxf(fminf(u1, 1.f), 1e-10f)) + 1e-10f);
            float l0 = p0[v] + b3[0] + g0;
            float l1 = p1[v] + b3[1] + g1;
            adj[idx] = (l1 > l0) ? 1.f : 0.f;     // argmax ties -> index 0 -> 0
        }
    }
}

// ---------------------------------------------------------------------------
// GAT attention scores: s1[i] = Wh[i]·a[:d], s2[i] = Wh[i]·a[d:]
// ---------------------------------------------------------------------------
__global__ void gat_scores_kernel(const float* __restrict__ Wh, int d,
                                  const float* __restrict__ a,
                                  float* __restrict__ s1, float* __restrict__ s2)
{
    int i = blockIdx.x * blockDim.x + threadIdx.x;
    if (i >= NA) return;
    const float* row = Wh + (size_t)i * d;
    float x1 = 0.f, x2 = 0.f;
    for (int k = 0; k < d; ++k) { x1 += row[k] * a[k]; x2 += row[k] * a[d + k]; }
    s1[i] = x1; s2[i] = x2;
}

// ---------------------------------------------------------------------------
// Masked row softmax: att[i,j] = softmax_j( adj>0 ? leaky(s1[i]+s2[j]) : -9e15 )
// One 256-thread block per row.
// ---------------------------------------------------------------------------
__global__ void att_softmax_kernel(const float* __restrict__ s1,
                                   const float* __restrict__ s2,
                                   const float* __restrict__ adj,
                                   float* __restrict__ att)
{
    __shared__ float red[256];
    const int i = blockIdx.x;
    const int tid = threadIdx.x;
    const float si = s1[i];
    const float* arow = adj + (size_t)i * NA;
    float* orow = att + (size_t)i * NA;

    float m = -3.4e38f;
    for (int j = tid; j < NA; j += 256) {
        float e = si + s2[j];
        e = (e > 0.f) ? e : 0.2f * e;                 // leaky_relu, slope 0.2
        float v = (arow[j] > 0.f) ? e : -9e15f;
        m = fmaxf(m, v);
    }
    red[tid] = m; __syncthreads();
    for (int s = 128; s > 0; s >>= 1) {
        if (tid < s) red[tid] = fmaxf(red[tid], red[tid + s]);
        __syncthreads();
    }
    m = red[0]; __syncthreads();

    float sum = 0.f;
    for (int j = tid; j < NA; j += 256) {
        float e = si + s2[j];
        e = (e > 0.f) ? e : 0.2f * e;
        float v = (arow[j] > 0.f) ? e : -9e15f;
        float ex = expf(v - m);
        orow[j] = ex;
        sum += ex;
    }
    red[tid] = sum; __syncthreads();
    for (int s = 128; s > 0; s >>= 1) {
        if (tid < s) red[tid] += red[tid + s];
        __syncthreads();
    }
    const float inv = 1.f / red[0];
    for (int j = tid; j < NA; j += 256) orow[j] *= inv;
}

// ---------------------------------------------------------------------------
// LSTM pointwise (gates already hold inp@Wih^T + h@Whh^T; biases added here)
// ---------------------------------------------------------------------------
__global__ void lstm_kernel(const float* __restrict__ gates,
                            const float* __restrict__ bih,
                            const float* __restrict__ bhh,
                            float* __restrict__ hidden, float* __restrict__ cell)
{
    int t = blockIdx.x * blockDim.x + threadIdx.x;
    if (t >= NA * HD) return;
    int n = t / HD, h = t - n * HD;
    const float* g = gates + (size_t)n * 4 * HD;
    float gi = g[h]          + bih[h]          + bhh[h];
    float gf = g[HD + h]     + bih[HD + h]     + bhh[HD + h];
    float gg = g[2 * HD + h] + bih[2 * HD + h] + bhh[2 * HD + h];
    float go = g[3 * HD + h] + bih[3 * HD + h] + bhh[3 * HD + h];
    float c2 = sigm_(gf) * cell[t] + sigm_(gi) * tanhf(gg);
    float h2 = sigm_(go) * tanhf(c2);
    cell[t] = c2;
    hidden[t] = h2;
}

// ---------------------------------------------------------------------------
// Policy/value head: action_mean, log_std broadcast, std=exp, value
// Output layout: mean[1024] | log_std[1024] | std[1024] | value[512] | h | c
// ---------------------------------------------------------------------------
__global__ void head_kernel(const float* __restrict__ hidden,
                            const float* __restrict__ actW,
                            const float* __restrict__ actB,
                            const float* __restrict__ logstd,
                            const float* __restrict__ valW,
                            const float* __restrict__ valB,
                            float* __restrict__ out)
{
    int n = blockIdx.x * blockDim.x + threadIdx.x;
    if (n >= NA) return;
    const float* h = hidden + (size_t)n * HD;
    float a0 = actB[0], a1 = actB[1], v = valB[0];
    for (int k = 0; k < HD; ++k) {
        float x = h[k];
        a0 += x * actW[k * 2];
        a1 += x * actW[k * 2 + 1];
        v  += x * valW[k];
    }
    out[n * 2]              = a0;
    out[n * 2 + 1]          = a1;
    out[NA * 2 + n * 2]     = logstd[0];
    out[NA * 2 + n * 2 + 1] = logstd[1];
    out[NA * 4 + n * 2]     = expf(logstd[0]);
    out[NA * 4 + n * 2 + 1] = expf(logstd[1]);
    out[NA * 6 + n]         = v;
}

// ---------------------------------------------------------------------------
extern "C" void kernel_launch(void* const* d_in, const int* in_sizes, int n_in,
                              void* d_out, int out_size, void* d_ws, size_t ws_size,
                              hipStream_t stream)
{
    (void)in_sizes; (void)n_in; (void)out_size; (void)ws_size;

    const float* x      = (const float*)d_in[0];
    const float* hid0   = (const float*)d_in[1];
    const float* cell0  = (const float*)d_in[2];
    const float* gu     = (const float*)d_in[3];
    const float* enc_W  = (const float*)d_in[4];
    const float* enc_b  = (const float*)d_in[5];
    const float* gat1_W = (const float*)d_in[6];
    const float* gat1_a = (const float*)d_in[7];
    const float* gat2_W = (const float*)d_in[8];
    const float* gat2_a = (const float*)d_in[9];
    const float* ha_W1  = (const float*)d_in[10];
    const float* ha_b1  = (const float*)d_in[11];
    const float* ha_W2  = (const float*)d_in[12];
    const float* ha_b2  = (const float*)d_in[13];
    const float* ha_W3  = (const float*)d_in[14];
    const float* ha_b3  = (const float*)d_in[15];
    const float* C_W    = (const float*)d_in[16];
    const float* C_b    = (const float*)d_in[17];
    const float* Wih    = (const float*)d_in[18];
    const float* Whh    = (const float*)d_in[19];
    const float* bih    = (const float*)d_in[20];
    const float* bhh    = (const float*)d_in[21];
    const float* act_W  = (const float*)d_in[22];
    const float* act_b  = (const float*)d_in[23];
    const float* logstd = (const float*)d_in[24];
    const float* val_W  = (const float*)d_in[25];
    const float* val_b  = (const float*)d_in[26];

    // workspace carve-out (floats), ~8.5 MB total
    float* ws     = (float*)d_ws;
    float* xs     = ws; ws += NA * HD;        // encoder output
    float* hidden = ws; ws += NA * HD;        // mutable state (never touch d_in)
    float* cell   = ws; ws += NA * HD;
    float* U      = ws; ws += NA * 128;       // hidden @ W1_top
    float* V      = ws; ws += NA * 128;       // hidden @ W1_bot
    float* adj    = ws; ws += NA * NA;
    float* att    = ws; ws += NA * NA;
    float* headWh = ws; ws += NA * HDH;       // per-head Wh (reused)
    float* comm   = ws; ws += NA * HD;        // head concat
    float* Wh2    = ws; ws += NA * HD;
    float* comm2  = ws; ws += NA * HD;        // gat2 output
    float* cbuf   = ws; ws += NA * HD;        // xs + comm2@C_W + C_b
    float* gates  = ws; ws += NA * 4 * HD;
    float* s1     = ws; ws += NA;
    float* s2     = ws; ws += NA;

    hipMemcpyAsync(hidden, hid0, NA * HD * sizeof(float), hipMemcpyDeviceToDevice, stream);
    hipMemcpyAsync(cell,  cell0, NA * HD * sizeof(float), hipMemcpyDeviceToDevice, stream);

    // xs = x @ enc_W + enc_b            (512x128)@(128x256)
    launch_gemm(x, NINP, enc_W, HD, 0, enc_b, xs, HD, NA, HD, NINP, 0, 0, stream);

    for (int p = 0; p < PASS; ++p) {
        // layer-1 decomposition of pairwise MLP: U = h@W1[:H], V = h@W1[H:]
        launch_gemm(hidden, HD, ha_W1,            128, 0, nullptr, U, 128, NA, 128, HD, 0, 0, stream);
        launch_gemm(hidden, HD, ha_W1 + HD * 128, 128, 0, nullptr, V, 128, NA, 128, HD, 0, 0, stream);

        // fused pair MLP + hard gumbel -> adjacency  (16384 waves)
        pair_adj_kernel<<<2048, 256, 0, stream>>>(U, V, ha_b1, ha_W2, ha_b2,
                                                  ha_W3, ha_b3,
                                                  gu + (size_t)p * NA * NA * 2, adj);

        // 4 GAT heads (concat=True -> ELU epilogue directly into comm slice)
        for (int k = 0; k < HEADS; ++k) {
            launch_gemm(hidden, HD, gat1_W + k * HD * HDH, HDH, 0, nullptr,
                        headWh, HDH, NA, HDH, HD, 0, 0, stream);
            gat_scores_kernel<<<2, 256, 0, stream>>>(headWh, HDH, gat1_a + k * 2 * HDH, s1, s2);
            att_softmax_kernel<<<NA, 256, 0, stream>>>(s1, s2, adj, att);
            launch_gemm(att, NA, headWh, HDH, 0, nullptr,
                        comm + k * HDH, HD, NA, HDH, NA, 0, /*ELU*/2, stream);
        }

        // aggregation GAT (concat=False, no activation)
        launch_gemm(comm, HD, gat2_W, HD, 0, nullptr, Wh2, HD, NA, HD, HD, 0, 0, stream);
        gat_scores_kernel<<<2, 256, 0, stream>>>(Wh2, HD, gat2_a, s1, s2);
        att_softmax_kernel<<<NA, 256, 0, stream>>>(s1, s2, adj, att);
        launch_gemm(att, NA, Wh2, HD, 0, nullptr, comm2, HD, NA, HD, NA, 0, 0, stream);

        // inp = xs + comm2 @ C_W[p] + C_b[p]   (accumulate into copy of xs)
        hipMemcpyAsync(cbuf, xs, NA * HD * sizeof(float), hipMemcpyDeviceToDevice, stream);
        launch_gemm(comm2, HD, C_W + p * HD * HD, HD, 0, C_b + p * HD,
                    cbuf, HD, NA, HD, HD, /*acc*/1, 0, stream);

        // gates = inp @ Wih^T + hidden @ Whh^T  (biases in pointwise kernel)
        launch_gemm(cbuf,   HD, Wih, HD, /*transB*/1, nullptr, gates, 4 * HD,
                    NA, 4 * HD, HD, 0, 0, stream);
        launch_gemm(hidden, HD, Whh, HD, /*transB*/1, nullptr, gates, 4 * HD,
                    NA, 4 * HD, HD, /*acc*/1, 0, stream);

        lstm_kernel<<<(NA * HD) / 256, 256, 0, stream>>>(gates, bih, bhh, hidden, cell);
    }

    float* out = (float*)d_out;
    head_kernel<<<2, 256, 0, stream>>>(hidden, act_W, act_b, logstd, val_W, val_b, out);
    // hidden at offset 3584, cell at 3584 + 131072
    hipMemcpyAsync(out + NA * 7,           hidden, NA * HD * sizeof(float),
                   hipMemcpyDeviceToDevice, stream);
    hipMemcpyAsync(out + NA * 7 + NA * HD, cell,   NA * HD * sizeof(float),
                   hipMemcpyDeviceToDevice, stream);
}